// Patch1DEmbedding_48507360641822
// MI455X (gfx1250) — compile-verified
//
#include <hip/hip_runtime.h>

typedef __attribute__((ext_vector_type(2))) float v2f;
typedef __attribute__((ext_vector_type(8))) float v8f;

#define B_      32
#define T_      512
#define D_      64
#define P_      16
#define DM_     512
#define STR_    8
#define NW_     63                    // (T - P)/STRIDE + 1
#define TROWS_  (D_ * NW_)            // 4032 temporal rows per batch
#define OUTROWS_ (TROWS_ + T_)        // 4544 rows per batch

// ---------------------------------------------------------------------------
// feature[b, t, m] = sum_d x[b,t,d] * Wd[m,d] + bd[m]
// GEMM: (B*T = 16384) x (DM = 512), K = 64.  One wave per 16x16 tile.
// ---------------------------------------------------------------------------
__global__ __launch_bounds__(256) void feature_kernel(
    const float* __restrict__ x, const float* __restrict__ Wd,
    const float* __restrict__ bd, float* __restrict__ out) {
  const int lane  = threadIdx.x & 31;
  const int wave  = blockIdx.x * (blockDim.x >> 5) + (threadIdx.x >> 5);
  const int col0  = (wave & 31) << 4;        // DM/16 = 32 col tiles
  const int row0  = (wave >> 5) << 4;        // B*T/16 = 1024 row tiles
  const int l15   = lane & 15;
  const int khalf = (lane >> 4) << 1;        // lanes 16..31 handle K+2,K+3

  const float* ax = x  + (size_t)(row0 + l15) * D_;   // A: row M = lane&15
  const float* bw = Wd + (size_t)(col0 + l15) * D_;   // B: col N = lane&15 (Wd row-major == B^T)

  v8f c = {};
#pragma unroll
  for (int k = 0; k < D_; k += 4) {
    v2f a, b;
    a.x = ax[k + khalf];      a.y = ax[k + khalf + 1];
    b.x = bw[k + khalf];      b.y = bw[k + khalf + 1];
    c = __builtin_amdgcn_wmma_f32_16x16x4_f32(false, a, false, b,
                                              (short)0, c, false, false);
  }

  const float bias = bd[col0 + l15];
  const int mrow_off = (lane >> 4) << 3;     // lanes 16..31 hold rows M+8
#pragma unroll
  for (int r = 0; r < 8; ++r) {
    const int grow = row0 + r + mrow_off;    // global (b*T + t)
    const int bi = grow >> 9;                // / T
    const int t  = grow & (T_ - 1);
    const size_t o = ((size_t)bi * OUTROWS_ + TROWS_ + t) * DM_ + col0 + l15;
    __builtin_nontemporal_store(c[r] + bias, out + o);
  }
}

// ---------------------------------------------------------------------------
// temporal[b, d*NW + w, m] = sum_p x[b, w*STR + p, d] * Wt[m,p] + bt[m]
// Per batch: GEMM (D*NW = 4032 rows) x (DM = 512), K = P = 16.
// Row tiles are 16 consecutive (d*NW + w) rows -> coalesced stores.
// ---------------------------------------------------------------------------
__global__ __launch_bounds__(256) void temporal_kernel(
    const float* __restrict__ x, const float* __restrict__ Wt,
    const float* __restrict__ bt, float* __restrict__ out) {
  const int lane  = threadIdx.x & 31;
  const int wave  = blockIdx.x * (blockDim.x >> 5) + (threadIdx.x >> 5);
  const int col0  = (wave & 31) << 4;        // 32 col tiles
  const int tmp   = wave >> 5;
  const int row0  = (tmp % (TROWS_ / 16)) << 4;   // 252 row tiles per batch
  const int bi    = tmp / (TROWS_ / 16);
  const int l15   = lane & 15;
  const int khalf = (lane >> 4) << 1;

  const int r = row0 + l15;                  // temporal row in [0, 4032)
  const int d = r / NW_;
  const int w = r - d * NW_;
  const float* ax = x  + ((size_t)bi * T_ + (size_t)w * STR_) * D_ + d;  // A[r][p] = x[b, w*8+p, d]
  const float* bw = Wt + (size_t)(col0 + l15) * P_;                      // B[p][m] = Wt[m*16+p]

  v8f c = {};
#pragma unroll
  for (int k = 0; k < P_; k += 4) {
    v2f a, b;
    a.x = ax[(size_t)(k + khalf) * D_];
    a.y = ax[(size_t)(k + khalf + 1) * D_];
    b.x = bw[k + khalf];
    b.y = bw[k + khalf + 1];
    c = __builtin_amdgcn_wmma_f32_16x16x4_f32(false, a, false, b,
                                              (short)0, c, false, false);
  }

  const float bias = bt[col0 + l15];
  const int mrow_off = (lane >> 4) << 3;
#pragma unroll
  for (int rr = 0; rr < 8; ++rr) {
    const int grow = row0 + rr + mrow_off;   // row within batch's temporal block
    const size_t o = ((size_t)bi * OUTROWS_ + grow) * DM_ + col0 + l15;
    __builtin_nontemporal_store(c[rr] + bias, out + o);
  }
}

extern "C" void kernel_launch(void* const* d_in, const int* in_sizes, int n_in,
                              void* d_out, int out_size, void* d_ws, size_t ws_size,
                              hipStream_t stream) {
  const float* x  = (const float*)d_in[0];
  const float* Wt = (const float*)d_in[1];
  const float* bt = (const float*)d_in[2];
  const float* Wd = (const float*)d_in[3];
  const float* bd = (const float*)d_in[4];
  float* out = (float*)d_out;

  // temporal: B * 252 row-tiles * 32 col-tiles = 258048 waves / 8 per block
  temporal_kernel<<<(B_ * (TROWS_ / 16) * (DM_ / 16)) / 8, 256, 0, stream>>>(x, Wt, bt, out);
  // feature: (B*T/16) * 32 = 32768 waves / 8 per block
  feature_kernel<<<((B_ * T_ / 16) * (DM_ / 16)) / 8, 256, 0, stream>>>(x, Wd, bd, out);
}